// SoftCountPixels_65910568124951
// MI455X (gfx1250) — compile-verified
//
#include <hip/hip_runtime.h>

typedef float v2f __attribute__((ext_vector_type(2)));
typedef float v8f __attribute__((ext_vector_type(8)));

// exp(-d / (2*GWIDTH2)) with GWIDTH2 = 0.3  ->  exp2(d * -log2(e)/0.6)
#define NEG_LOG2E_OVER_0P6 (-2.4044917348148930f)

__device__ __forceinline__ float soft_exp(float dist2) {
    // |dist2|: free source modifier folded into v_sqrt_f32. dist2 < 0 only via
    // cancellation when true dist ~ 0, where |dist2| is tiny and exp(...) ~ 1
    // -- the correct value. No clamp instructions needed.
    float t = __builtin_amdgcn_sqrtf(__builtin_fabsf(dist2)); // v_sqrt_f32 |s| (TRANS)
    return __builtin_amdgcn_exp2f(t * NEG_LOG2E_OVER_0P6);    // v_exp_f32 (TRANS)
}

// One wave processes pixPerWave consecutive pixels of one batch image against all
// 32 prototypes using V_WMMA_F32_16X16X4_F32 tiles.
__global__ __launch_bounds__(256) void SoftCountPixels_wmma_kernel(
    const float* __restrict__ x,   // (b, 3, hw) channel-planar
    const float* __restrict__ P,   // (32, 3)
    float* __restrict__ out,       // (b, 32), pre-zeroed
    int hw, int wavesPerBatch, int pixPerWave)
{
    const int lane       = threadIdx.x & 31;
    const int waveInBlk  = threadIdx.x >> 5;
    const int wavesBlk   = blockDim.x >> 5;
    const int waveId     = blockIdx.x * wavesBlk + waveInBlk;
    const int b          = waveId / wavesPerBatch;
    const int chunk      = waveId % wavesPerBatch;
    const int pixBase    = chunk * pixPerWave;

    const float* __restrict__ xr = x + (size_t)b * 3 * hw;
    const float* __restrict__ xg = xr + hw;
    const float* __restrict__ xb = xg + hw;

    // ---- Build B (4x16, K = {-2Pr, -2Pg, -2Pb, 1}) and C (= |P|^2 broadcast) ----
    // B layout mirrors the documented 16x4 f32 A layout:
    //   VGPR0: lanes 0-15 hold K=0, lanes 16-31 hold K=2
    //   VGPR1: lanes 0-15 hold K=1, lanes 16-31 hold K=3
    const int  n0 = lane & 15;       // prototype column for N-tile 0
    const int  n1 = n0 + 16;         // prototype column for N-tile 1
    const bool lo = (lane < 16);

    float p0r = P[n0 * 3 + 0], p0g = P[n0 * 3 + 1], p0b = P[n0 * 3 + 2];
    float p1r = P[n1 * 3 + 0], p1g = P[n1 * 3 + 1], p1b = P[n1 * 3 + 2];
    float psq0 = p0r * p0r + p0g * p0g + p0b * p0b;
    float psq1 = p1r * p1r + p1g * p1g + p1b * p1b;

    v2f B0, B1;
    B0.x = lo ? (-2.0f * p0r) : (-2.0f * p0b);
    B0.y = lo ? (-2.0f * p0g) : 1.0f;
    B1.x = lo ? (-2.0f * p1r) : (-2.0f * p1b);
    B1.y = lo ? (-2.0f * p1g) : 1.0f;

    v8f C0, C1;
#pragma unroll
    for (int k = 0; k < 8; ++k) { C0[k] = psq0; C1[k] = psq1; }

    v8f acc0 = {};  // prototypes  0..15 accumulator (C/D layout)
    v8f acc1 = {};  // prototypes 16..31 accumulator

    // ---- Main loop: 32 pixels / iteration = two 16-row A tiles ----
    for (int it = 0; it < pixPerWave; it += 32) {
        const int i = pixBase + it + lane;

        float r  = xr[i];
        float g  = xg[i];
        float bl = xb[i];
        // prefetch next iteration's cachelines (global_prefetch_b8)
        __builtin_prefetch(xr + i + 32, 0, 1);
        __builtin_prefetch(xg + i + 32, 0, 1);
        __builtin_prefetch(xb + i + 32, 0, 1);

        float s = r * r + g * g + bl * bl;   // |x|^2 (fma chain)

        // half-swap copies for building the A tiles
        float r2 = __shfl_xor(r,  16, 32);
        float g2 = __shfl_xor(g,  16, 32);
        float b2 = __shfl_xor(bl, 16, 32);
        float s2 = __shfl_xor(s,  16, 32);

        // A tile 0 = pixels [base .. base+15]:
        //   lanes 0-15 need (r,g) of own pixel, lanes 16-31 need (b,|x|^2) of pixel lane-16
        v2f A0, A1;
        A0.x = lo ? r  : b2;
        A0.y = lo ? g  : s2;
        // A tile 1 = pixels [base+16 .. base+31]
        A1.x = lo ? r2 : bl;
        A1.y = lo ? g2 : s;

        // dist^2 = |x|^2 - 2 x.P + |P|^2  via one WMMA per (pixel-tile, proto-tile)
        v8f d00 = __builtin_amdgcn_wmma_f32_16x16x4_f32(false, A0, false, B0, (short)0, C0, false, false);
        v8f d01 = __builtin_amdgcn_wmma_f32_16x16x4_f32(false, A0, false, B1, (short)0, C1, false, false);
        v8f d10 = __builtin_amdgcn_wmma_f32_16x16x4_f32(false, A1, false, B0, (short)0, C0, false, false);
        v8f d11 = __builtin_amdgcn_wmma_f32_16x16x4_f32(false, A1, false, B1, (short)0, C1, false, false);

#pragma unroll
        for (int k = 0; k < 8; ++k) {
            acc0[k] += soft_exp(d00[k]);
            acc0[k] += soft_exp(d10[k]);
            acc1[k] += soft_exp(d01[k]);
            acc1[k] += soft_exp(d11[k]);
        }
    }

    // ---- Reduce: sum 8 M-rows per lane, then fold lanes l and l^16 (same N) ----
    float s0 = acc0[0] + acc0[1] + acc0[2] + acc0[3] + acc0[4] + acc0[5] + acc0[6] + acc0[7];
    float s1 = acc1[0] + acc1[1] + acc1[2] + acc1[3] + acc1[4] + acc1[5] + acc1[6] + acc1[7];
    s0 += __shfl_xor(s0, 16, 32);
    s1 += __shfl_xor(s1, 16, 32);

    const float scale = 1.0f / (float)hw;
    if (lo) {
        atomicAdd(out + b * 32 + n0, s0 * scale);
        atomicAdd(out + b * 32 + n1, s1 * scale);
    }
}

__global__ void SoftCountPixels_zero_kernel(float* __restrict__ out, int n) {
    int i = blockIdx.x * blockDim.x + threadIdx.x;
    if (i < n) out[i] = 0.0f;
}

extern "C" void kernel_launch(void* const* d_in, const int* in_sizes, int n_in,
                              void* d_out, int out_size, void* d_ws, size_t ws_size,
                              hipStream_t stream) {
    const float* x = (const float*)d_in[0];
    const float* P = (const float*)d_in[1];
    float* out = (float*)d_out;

    const int hw      = 256 * 256;                 // per reference
    const int batches = in_sizes[0] / (3 * hw);    // = 16

    // harness poisons d_out; zero it (atomics accumulate into it)
    SoftCountPixels_zero_kernel<<<(out_size + 255) / 256, 256, 0, stream>>>(out, out_size);

    const int wavesPerBatch = 64;                  // 1024 pixels per wave
    const int pixPerWave    = hw / wavesPerBatch;  // 1024 -> 32 WMMA macro-iters
    const int threads       = 256;                 // 8 waves per block
    const int totalWaves    = batches * wavesPerBatch;
    const int blocks        = totalWaves / (threads / 32);

    SoftCountPixels_wmma_kernel<<<blocks, threads, 0, stream>>>(
        x, P, out, hw, wavesPerBatch, pixPerWave);
}